// LearnPermutations_68934225101553
// MI455X (gfx1250) — compile-verified
//
#include <hip/hip_runtime.h>
#include <math.h>

#define Nmat   256
#define KPERM  256
#define ELEMS  (Nmat * Nmat)     // 65536 elements = 256 KB f32
#define TPB    1024              // 32 wave32 waves per workgroup
#define VPW    (ELEMS / (4 * TPB))  // 16 float4 per thread

typedef unsigned int u32;
typedef u32   __attribute__((ext_vector_type(4))) v4u;
typedef int   __attribute__((ext_vector_type(4))) v4i;
typedef int   __attribute__((ext_vector_type(8))) v8i;
typedef float __attribute__((ext_vector_type(4))) v4f;   // native vector: OK for NT builtins

#if defined(__has_builtin)
#  if __has_builtin(__builtin_amdgcn_tensor_load_to_lds)
#    define HAVE_TDM 1
#  endif
#endif

__device__ __forceinline__ float waveMax32(float v) {
#pragma unroll
  for (int o = 16; o > 0; o >>= 1) v = fmaxf(v, __shfl_xor(v, o, 32));
  return v;
}
__device__ __forceinline__ float waveSum32(float v) {
#pragma unroll
  for (int o = 16; o > 0; o >>= 1) v += __shfl_xor(v, o, 32);
  return v;
}
__device__ __forceinline__ float gumbel(float uu) {
  return -__logf(-__logf(uu + 1e-20f) + 1e-20f) * 0.01f;
}

__global__ __launch_bounds__(TPB)
void LearnPermutations_sinkhorn_kernel(const float* __restrict__ plw,
                                       const float* __restrict__ uin,
                                       const int*   __restrict__ iterp,
                                       float*       __restrict__ out) {
  extern __shared__ float lds[];
  float* Msh  = lds;                    // [256*256] the matrix, LDS-resident
  float* sMax = lds + ELEMS;            // [4*256] column partial maxima
  float* sSum = sMax + 4 * Nmat;        // [4*256] column partial sums
  float* cLse = sSum + 4 * Nmat;        // [256]  pending column logsumexp

  const int tid  = threadIdx.x;
  const int k    = blockIdx.x;
  const size_t base = (size_t)k * ELEMS;

  // ---- schedule scalars (device input; double math to match numpy exactly,
  //      n_iters sits on an int-truncation cliff so f32 here would be unsafe) ----
  double frac = (double)iterp[0] / 100000.0;
  frac = frac < 0.0 ? 0.0 : (frac > 1.0 ? 1.0 : frac);
  const float tau     = (float)pow(10.0, -3.0 + frac * (-4.0)); // log10 interp
  const int   n_iters = (int)(20.0 + frac * 130.0);

  // ---- CDNA5 TDM: DMA plw[k] (256x256 f32 tile) from HBM into LDS ----
#if HAVE_TDM
  if (tid < 32) {  // one wave issues the descriptor
    unsigned long long ga = (unsigned long long)(const void*)(plw + base);
    u32 ldsOff = (u32)(unsigned long long)(void*)Msh;  // low 32 bits = LDS byte offset
    v4u g0;
    g0.x = 1u;                                         // count=1 (valid descriptor)
    g0.y = ldsOff;                                     // lds_addr
    g0.z = (u32)ga;                                    // global_addr[31:0]
    g0.w = ((u32)(ga >> 32) & 0x01FFFFFFu) | 0x80000000u; // addr[56:32] | type=2
    v8i g1;
    g1[0] = 0x00020000;      // wg_mask=0, data_size=2 (4B), no flags
    g1[1] = 0x01000000;      // tensor_dim0=256 (low16 in [31:16])
    g1[2] = 0x01000000;      // tensor_dim0 hi=0 | tensor_dim1=256 low16
    g1[3] = 0x01000000;      // tensor_dim1 hi=0 | tile_dim0=256
    g1[4] = 0x00000100;      // tile_dim1=256 | tile_dim2=0
    g1[5] = 256;             // tensor_dim0_stride = 256 elements
    g1[6] = 0;               // stride hi | tensor_dim1_stride lo (unused, 2D)
    g1[7] = 0;
    v4i z4 = {0, 0, 0, 0};
#  if __clang_major__ >= 23
    v8i z8 = {0, 0, 0, 0, 0, 0, 0, 0};
    __builtin_amdgcn_tensor_load_to_lds(g0, g1, z4, z4, z8, 0);
#  else
    __builtin_amdgcn_tensor_load_to_lds(g0, g1, z4, z4, 0);
#  endif
  }
  // overlap: warm caches with u while TDM streams plw into LDS
#pragma unroll
  for (int c = 0; c < VPW; c += 4)
    __builtin_prefetch(&uin[base + 4 * (tid + c * TPB)], 0, 1);
  if (tid < 32) __builtin_amdgcn_s_wait_tensorcnt(0);
#else
  {
    const v4f* p4 = (const v4f*)(plw + base);
    v4f* m4 = (v4f*)Msh;
#pragma unroll 4
    for (int c = 0; c < VPW; ++c) {
      int i4 = tid + c * TPB;
      m4[i4] = __builtin_nontemporal_load(&p4[i4]);
    }
  }
#endif
  __syncthreads();

  // ---- fold Gumbel noise + temperature in place: M = (plw + noise)/tau ----
  {
    const v4f* u4 = (const v4f*)(uin + base);
#pragma unroll 4
    for (int c = 0; c < VPW; ++c) {
      int i4 = tid + c * TPB;               // b128-coalesced, read-once -> NT
      v4f uu = __builtin_nontemporal_load(&u4[i4]);
      v4f mm = *(const v4f*)&Msh[4 * i4];
      mm.x = (mm.x + gumbel(uu.x)) / tau;
      mm.y = (mm.y + gumbel(uu.y)) / tau;
      mm.z = (mm.z + gumbel(uu.z)) / tau;
      mm.w = (mm.w + gumbel(uu.w)) / tau;
      *(v4f*)&Msh[4 * i4] = mm;
    }
  }
  if (tid < Nmat) cLse[tid] = 0.0f;
  __syncthreads();

  const int wave  = tid >> 5;
  const int lane  = tid & 31;
  const int jcol  = tid & (Nmat - 1);   // column owned in the column pass
  const int chunk = tid >> 8;           // row chunk 0..3 (64 rows each)
  const int rbeg  = chunk * 64;

  for (int it = 0; it < n_iters; ++it) {
    // ---- row pass: apply pending column LSE, then row-normalize ----
    // wave w owns rows [8w, 8w+8); lanes cover 32 consecutive cols -> no bank conflicts
    for (int rr = 0; rr < 8; ++rr) {
      const int r = wave * 8 + rr;
      float x[8];
      float m = -INFINITY;
#pragma unroll
      for (int c = 0; c < 8; ++c) {
        const int j = lane + 32 * c;
        x[c] = Msh[r * Nmat + j] - cLse[j];
        m = fmaxf(m, x[c]);
      }
      m = waveMax32(m);
      float s = 0.0f;
#pragma unroll
      for (int c = 0; c < 8; ++c) s += __expf(x[c] - m);
      s = waveSum32(s);
      const float lse = m + __logf(s);
#pragma unroll
      for (int c = 0; c < 8; ++c) Msh[r * Nmat + lane + 32 * c] = x[c] - lse;
    }
    __syncthreads();

    // ---- column pass: compute colLse[j], defer the subtraction ----
    float m = -INFINITY;
#pragma unroll 8
    for (int i = 0; i < 64; ++i) m = fmaxf(m, Msh[(rbeg + i) * Nmat + jcol]);
    sMax[chunk * Nmat + jcol] = m;
    __syncthreads();
    const float gm = fmaxf(fmaxf(sMax[jcol], sMax[Nmat + jcol]),
                           fmaxf(sMax[2 * Nmat + jcol], sMax[3 * Nmat + jcol]));
    float s = 0.0f;
#pragma unroll 8
    for (int i = 0; i < 64; ++i) s += __expf(Msh[(rbeg + i) * Nmat + jcol] - gm);
    sSum[chunk * Nmat + jcol] = s;
    __syncthreads();
    if (chunk == 0) {
      const float tot = sSum[jcol] + sSum[Nmat + jcol] +
                        sSum[2 * Nmat + jcol] + sSum[3 * Nmat + jcol];
      cLse[jcol] = gm + __logf(tot);
    }
    __syncthreads();
  }

  // ---- output: exp with last pending column LSE applied; NT b128 stream-out ----
  {
    v4f* o4 = (v4f*)(out + base);
#pragma unroll 4
    for (int c = 0; c < VPW; ++c) {
      int i4 = tid + c * TPB;
      int e  = 4 * i4;
      int j  = e & (Nmat - 1);            // 4 consecutive columns, no wrap (4|256)
      v4f mm = *(const v4f*)&Msh[e];
      v4f r;
      r.x = __expf(mm.x - cLse[j + 0]);
      r.y = __expf(mm.y - cLse[j + 1]);
      r.z = __expf(mm.z - cLse[j + 2]);
      r.w = __expf(mm.w - cLse[j + 3]);
      __builtin_nontemporal_store(r, &o4[i4]);
    }
  }
}

extern "C" void kernel_launch(void* const* d_in, const int* in_sizes, int n_in,
                              void* d_out, int out_size, void* d_ws, size_t ws_size,
                              hipStream_t stream) {
  const float* plw   = (const float*)d_in[0];
  const float* u     = (const float*)d_in[1];
  const int*   iters = (const int*)d_in[2];
  float*       out   = (float*)d_out;
  (void)in_sizes; (void)n_in; (void)out_size; (void)d_ws; (void)ws_size;

  const size_t smem = (size_t)(ELEMS + 4 * Nmat + 4 * Nmat + Nmat) * sizeof(float); // ~265 KB < 320 KB/WGP
  LearnPermutations_sinkhorn_kernel<<<dim3(KPERM), dim3(TPB), smem, stream>>>(plw, u, iters, out);
}